// LM_Softmax_32615981645893
// MI455X (gfx1250) — compile-verified
//
#include <hip/hip_runtime.h>
#include <hip/hip_bf16.h>
#include <cstdint>

// ---------------------------------------------------------------------------
// LM margin-softmax loss, MI455X (gfx1250).
// Memory-bound: 823 MB read once -> ~35us floor @ 23.3 TB/s. No matmul
// structure, so no WMMA; instead we use the CDNA5 async global->LDS mover
// (GLOBAL_LOAD_ASYNC_TO_LDS_B128, ASYNCcnt) to double-buffer row chunks and
// overlap HBM streaming with the exp/max passes. wave32 shuffles reduce.
// ---------------------------------------------------------------------------

#define CHUNK  4096   // floats per LDS chunk (16 KB)
#define BLOCK  256    // threads per block (8 wave32)
#define V4PT   4      // float4 per thread per chunk = CHUNK/4/BLOCK

// One async copy: 16 bytes/lane, global -> LDS, tracked by ASYNCcnt.
// Low 32 bits of a generic LDS pointer are the workgroup-relative LDS byte
// address, which is exactly what the instruction's VDST operand wants.
__device__ __forceinline__ void async_ld16(float* lds_dst, const float* gsrc) {
    uint32_t loff = (uint32_t)(uintptr_t)lds_dst;
    asm volatile("global_load_async_to_lds_b128 %0, %1, off"
                 :: "v"(loff), "v"(gsrc)
                 : "memory");
}

__device__ __forceinline__ void issue_chunk(float* ldsbuf, const float* gchunk, int tid) {
    #pragma unroll
    for (int j = 0; j < V4PT; ++j) {
        int f4 = tid + j * BLOCK;                 // float4 index within chunk
        async_ld16(ldsbuf + 4 * f4, gchunk + 4 * f4);
    }
}

extern "C" __global__ __launch_bounds__(BLOCK)
void lmx_row_kernel(const float* __restrict__ src, const int* __restrict__ lab,
                    float* __restrict__ partial, int C) {
    extern __shared__ float smem[];               // 2*CHUNK buffers + 16 scratch
    const int tid = threadIdx.x;
    const int row = blockIdx.x;
    const float* rowp = src + (size_t)row * (size_t)C;

    // Margin-modified label logit (raw l decides the branch).
    const int   label = lab[row];
    const float l     = rowp[label];              // broadcast load, one line
    const float scal  = l * 1.1f;                 // l * (1 + m)
    const float val   = (l > 0.0f) ? (scal / 1.05f) : (scal * 1.05f);

    const int full     = C / CHUNK;               // 12 full chunks for C=50257
    const int base_rem = full * CHUNK;

    float m = -INFINITY;                          // online softmax state
    float s = 0.0f;

    float* pcur  = smem;
    float* pnext = smem + CHUNK;

    if (full > 0) issue_chunk(pcur, rowp, tid);   // prologue: chunk 0 in flight

    for (int k = 0; k < full; ++k) {
        if (k + 1 < full) {
            issue_chunk(pnext, rowp + (size_t)(k + 1) * CHUNK, tid);
            // 4 newer loads outstanding -> chunk k has fully landed.
            asm volatile("s_wait_asynccnt 4" ::: "memory");
        } else {
            asm volatile("s_wait_asynccnt 0" ::: "memory");
        }
        __syncthreads();                          // all waves' writes visible

        if (label / CHUNK == k) {                 // uniform branch
            if (tid == 0) pcur[label - k * CHUNK] = val;
            __syncthreads();
        }

        const float4* b4 = (const float4*)pcur;
        float4 a = b4[tid];
        float4 b = b4[tid + BLOCK];
        float4 c = b4[tid + 2 * BLOCK];
        float4 d = b4[tid + 3 * BLOCK];

        float lm = fmaxf(fmaxf(fmaxf(fmaxf(a.x, a.y), fmaxf(a.z, a.w)),
                               fmaxf(fmaxf(b.x, b.y), fmaxf(b.z, b.w))),
                         fmaxf(fmaxf(fmaxf(c.x, c.y), fmaxf(c.z, c.w)),
                               fmaxf(fmaxf(d.x, d.y), fmaxf(d.z, d.w))));
        float nm = fmaxf(m, lm);
        s *= __expf(m - nm);                      // exp(-inf)=0 handles first pass
        s += __expf(a.x - nm) + __expf(a.y - nm) + __expf(a.z - nm) + __expf(a.w - nm)
           + __expf(b.x - nm) + __expf(b.y - nm) + __expf(b.z - nm) + __expf(b.w - nm)
           + __expf(c.x - nm) + __expf(c.y - nm) + __expf(c.z - nm) + __expf(c.w - nm)
           + __expf(d.x - nm) + __expf(d.y - nm) + __expf(d.z - nm) + __expf(d.w - nm);
        m = nm;

        __syncthreads();                          // done reading pcur -> reusable
        float* t = pcur; pcur = pnext; pnext = t;
    }

    // Remainder (C % CHUNK = 1105 elements): scalar loads straight from global.
    for (int idx = base_rem + tid; idx < C; idx += BLOCK) {
        float v = rowp[idx];
        if (idx == label) v = val;
        float nm = fmaxf(m, v);
        s = s * __expf(m - nm) + __expf(v - nm);
        m = nm;
    }

    // wave32 butterfly combine of (m, s) pairs.
    #pragma unroll
    for (int off = 16; off > 0; off >>= 1) {
        float mo = __shfl_xor(m, off, 32);
        float so = __shfl_xor(s, off, 32);
        float nm = fmaxf(m, mo);
        s = s * __expf(m - nm) + so * __expf(mo - nm);
        m = nm;
    }

    float* smx = smem + 2 * CHUNK;                // 8 maxes
    float* ssx = smx + (BLOCK / 32);              // 8 sums
    if ((tid & 31) == 0) { smx[tid >> 5] = m; ssx[tid >> 5] = s; }
    __syncthreads();
    if (tid == 0) {
        float M = smx[0], S = ssx[0];
        #pragma unroll
        for (int w = 1; w < BLOCK / 32; ++w) {
            float mo = smx[w], so = ssx[w];
            float nm = fmaxf(M, mo);
            S = S * __expf(M - nm) + so * __expf(mo - nm);
            M = nm;
        }
        // -logp[label] = -(val - (M + log S))
        partial[row] = (M + __logf(S)) - val;
    }
}

extern "C" __global__ __launch_bounds__(256)
void lmx_reduce_kernel(const float* __restrict__ partial, float* __restrict__ out, int N) {
    __shared__ float red[8];
    const int tid = threadIdx.x;
    float s = 0.0f;
    for (int i = tid; i < N; i += 256) s += partial[i];   // fixed order: deterministic
    #pragma unroll
    for (int off = 16; off > 0; off >>= 1) s += __shfl_xor(s, off, 32);
    if ((tid & 31) == 0) red[tid >> 5] = s;
    __syncthreads();
    if (tid == 0) {
        float t = 0.0f;
        #pragma unroll
        for (int w = 0; w < 8; ++w) t += red[w];
        out[0] = t / (float)N;
    }
}

extern "C" void kernel_launch(void* const* d_in, const int* in_sizes, int n_in,
                              void* d_out, int out_size, void* d_ws, size_t ws_size,
                              hipStream_t stream) {
    const float* src = (const float*)d_in[0];     // [N, C] f32
    const int*   lab = (const int*)d_in[1];       // [N] int
    const int N = in_sizes[1];
    const int C = in_sizes[0] / N;                // 50257

    float* partial = (float*)d_ws;                // N floats of scratch

    const size_t shmem = (size_t)(2 * CHUNK + 16) * sizeof(float);  // 32,832 B
    lmx_row_kernel<<<N, BLOCK, shmem, stream>>>(src, lab, partial, C);
    lmx_reduce_kernel<<<1, 256, 0, stream>>>(partial, (float*)d_out, N);
}